// Attention_55722905698733
// MI455X (gfx1250) — compile-verified
//
#include <hip/hip_runtime.h>
#include <math.h>

typedef __attribute__((ext_vector_type(2))) float v2f;
typedef __attribute__((ext_vector_type(8))) float v8f;

#define N_ 64
#define T_ 2048
#define D_ 512
#define NEG_ (-1e9f)
#define NDS_ ((size_t)N_ * D_)   // row stride (elements) in key/value

// -----------------------------------------------------------------------------
// Kernel 1 (templated on slice count): per (batch n, slice s):
//   Stage A: e[t] = K[t,n,:].q[n,:]      via V_WMMA_F32_16X16X4_F32
//   masked slice-local softmax stats (max, sum of exp)
//   Stage B: ctx_s[d] = sum_t p[t]V[t,n,d] via V_WMMA_F32_16X16X4_F32
// Block size == slice length TC so the softmax stage is one t per thread.
// -----------------------------------------------------------------------------
template <int SLICES>
__global__ __launch_bounds__(T_ / SLICES) void attn_partial_kernel(
    const float* __restrict__ q,    // (N, D)
    const float* __restrict__ k,    // (T, N, D)
    const float* __restrict__ v,    // (T, N, D)
    const int*   __restrict__ lens, // (N)
    float* __restrict__ ws_m,       // (N*SLICES)
    float* __restrict__ ws_sum,     // (N*SLICES)
    float* __restrict__ ws_ctx)     // (N*SLICES*D) unnormalized partial ctx
{
    constexpr int TC    = T_ / SLICES;  // timesteps per slice
    constexpr int BLOCK = TC;           // threads per block
    constexpr int NW    = BLOCK / 32;   // waves per block

    __shared__ float lds_q[D_];
    __shared__ float lds_e[TC];         // energies, later softmax weights p
    __shared__ float lds_red[BLOCK];

    const int n    = blockIdx.x;
    const int s    = blockIdx.y;
    const int t0s  = s * TC;
    const int tid  = threadIdx.x;
    const int lane = tid & 31;
    const int wave = tid >> 5;
    const int half = lane >> 4;         // 0: k-pair {0,1}, 1: k-pair {2,3}
    const int lm16 = lane & 15;

    for (int i = tid; i < D_; i += BLOCK) lds_q[i] = q[n * D_ + i];
    __syncthreads();

    // ---- Stage A: energies (A = 16 t-rows x 4 d, B = q broadcast) ----
    for (int tile = wave; tile < TC / 16; tile += NW) {
        const int tbase = t0s + tile * 16;
        const float* krow =
            k + ((size_t)(tbase + lm16) * N_ + n) * D_ + 2 * half;
        v8f acc = {};
#pragma unroll 8
        for (int d = 0; d < D_; d += 4) {
            if ((d & 31) == 0)                       // once per 8 WMMAs
                __builtin_prefetch(krow + d + 128, 0, 0);  // 512B/row ahead
            v2f a, b;
            a.x = krow[d + 0];                 // A[m=lm16, k=2*half]
            a.y = krow[d + 1];                 // A[m=lm16, k=2*half+1]
            b.x = lds_q[d + 2 * half + 0];     // B[k=2*half,   all cols]
            b.y = lds_q[d + 2 * half + 1];     // B[k=2*half+1, all cols]
            acc = __builtin_amdgcn_wmma_f32_16x16x4_f32(
                false, a, false, b, (short)0, acc, false, false);
        }
        // columns identical; column 0: lane 0 has rows 0..7, lane 16 rows 8..15
        if (lane == 0) {
#pragma unroll
            for (int r = 0; r < 8; ++r) lds_e[tile * 16 + r] = acc[r];
        } else if (lane == 16) {
#pragma unroll
            for (int r = 0; r < 8; ++r) lds_e[tile * 16 + 8 + r] = acc[r];
        }
    }
    __syncthreads();

    // ---- masked slice softmax stats (tid == local t) ----
    const int len = lens[n];
    float ev = lds_e[tid];
    if (t0s + tid >= len) ev = NEG_;

    lds_red[tid] = ev;
    __syncthreads();
    for (int off = BLOCK / 2; off > 0; off >>= 1) {
        if (tid < off) lds_red[tid] = fmaxf(lds_red[tid], lds_red[tid + off]);
        __syncthreads();
    }
    const float mslice = lds_red[0];
    __syncthreads();

    const float p = __expf(ev - mslice);
    lds_e[tid]   = p;
    lds_red[tid] = p;
    __syncthreads();
    for (int off = BLOCK / 2; off > 0; off >>= 1) {
        if (tid < off) lds_red[tid] += lds_red[tid + off];
        __syncthreads();
    }
    if (tid == 0) {
        ws_m[n * SLICES + s]   = mslice;
        ws_sum[n * SLICES + s] = lds_red[0];
    }
    __syncthreads();

    // ---- Stage B: partial context, 32-wide d-blocks (full 128B lines/row) ----
    float* ctx_out = ws_ctx + ((size_t)(n * SLICES + s)) * D_;
    for (int db = wave; db < D_ / 32; db += NW) {
        const int d0 = db * 32;
        const float* vcol =
            v + ((size_t)(t0s + 2 * half) * N_ + n) * D_ + d0 + lm16;
        v8f acc0 = {}, acc1 = {};
#pragma unroll 2
        for (int tb = 0; tb < TC; tb += 4) {
            const float* vp = vcol + (size_t)tb * NDS_;
            __builtin_prefetch(vp + 16 * NDS_, 0, 0);  // rows tb+16 / tb+18
            __builtin_prefetch(vp + 17 * NDS_, 0, 0);  // rows tb+17 / tb+19
            v2f a0, a1, b;
            b.x  = lds_e[tb + 2 * half + 0];
            b.y  = lds_e[tb + 2 * half + 1];
            a0.x = vp[0];                  // V[t0s+tb+2h  ][n][d0+lm16]
            a0.y = vp[NDS_];               // V[t0s+tb+2h+1][n][d0+lm16]
            a1.x = vp[16];                 // same rows, d0+16+lm16
            a1.y = vp[NDS_ + 16];
            acc0 = __builtin_amdgcn_wmma_f32_16x16x4_f32(
                false, a0, false, b, (short)0, acc0, false, false);
            acc1 = __builtin_amdgcn_wmma_f32_16x16x4_f32(
                false, a1, false, b, (short)0, acc1, false, false);
        }
        if (lane == 0) {
#pragma unroll
            for (int r = 0; r < 8; ++r) {
                ctx_out[d0 + r]      = acc0[r];
                ctx_out[d0 + 16 + r] = acc1[r];
            }
        } else if (lane == 16) {
#pragma unroll
            for (int r = 0; r < 8; ++r) {
                ctx_out[d0 + 8 + r]  = acc0[r];
                ctx_out[d0 + 24 + r] = acc1[r];
            }
        }
    }
}

// -----------------------------------------------------------------------------
// Kernel 2: combine SLICES per batch via log-sum-exp and normalize.
// -----------------------------------------------------------------------------
template <int SLICES>
__global__ __launch_bounds__(256) void attn_combine_kernel(
    const float* __restrict__ ws_m,
    const float* __restrict__ ws_sum,
    const float* __restrict__ ws_ctx,
    float* __restrict__ out)      // (N, D)
{
    const int n   = blockIdx.x;
    const int tid = threadIdx.x;

    float m[SLICES], sm[SLICES], scale[SLICES];
    float M = -INFINITY;
#pragma unroll
    for (int s = 0; s < SLICES; ++s) {
        m[s]  = ws_m[n * SLICES + s];
        sm[s] = ws_sum[n * SLICES + s];
        M = fmaxf(M, m[s]);
    }
    float total = 0.f;
#pragma unroll
    for (int s = 0; s < SLICES; ++s) {
        scale[s] = __expf(m[s] - M);
        total += scale[s] * sm[s];
    }
    const float inv = 1.0f / total;

    for (int d = tid; d < D_; d += 256) {
        float c = 0.f;
#pragma unroll
        for (int s = 0; s < SLICES; ++s)
            c += scale[s] * ws_ctx[((size_t)(n * SLICES + s)) * D_ + d];
        out[n * D_ + d] = c * inv;
    }
}

template <int SLICES>
static void launch_impl(const float* q, const float* k, const float* v,
                        const int* lens, float* out, float* ws,
                        hipStream_t stream) {
    float* ws_m   = ws;
    float* ws_sum = ws + N_ * SLICES;
    float* ws_ctx = ws + 2 * N_ * SLICES;
    dim3 grid1(N_, SLICES);
    attn_partial_kernel<SLICES><<<grid1, T_ / SLICES, 0, stream>>>(
        q, k, v, lens, ws_m, ws_sum, ws_ctx);
    attn_combine_kernel<SLICES><<<N_, 256, 0, stream>>>(ws_m, ws_sum, ws_ctx,
                                                        out);
}

extern "C" void kernel_launch(void* const* d_in, const int* in_sizes, int n_in,
                              void* d_out, int out_size, void* d_ws, size_t ws_size,
                              hipStream_t stream) {
    (void)in_sizes; (void)n_in; (void)out_size;
    const float* q    = (const float*)d_in[0];
    const float* k    = (const float*)d_in[1];
    const float* v    = (const float*)d_in[2];
    const int*   lens = (const int*)d_in[3];
    float* out = (float*)d_out;
    float* ws  = (float*)d_ws;

    const size_t need16 = (size_t)(2 * N_ * 16 + N_ * 16 * D_) * sizeof(float);
    if (ws_size >= need16)
        launch_impl<16>(q, k, v, lens, out, ws, stream);  // 1024 WGs
    else
        launch_impl<8>(q, k, v, lens, out, ws, stream);   // 512 WGs, ~1 MB ws
}